// FocalLoss_31954556682973
// MI455X (gfx1250) — compile-verified
//
#include <hip/hip_runtime.h>
#include <hip/hip_bf16.h>
#include <math.h>

// ---------------------------------------------------------------------------
// RetinaNet focal loss for MI455X (gfx1250, wave32).
//   pass1: one block = (image, anchor-chunk). Annotations broadcast to LDS via
//          async global->LDS b128 copies (ASYNCcnt), per-anchor IoU argmax with
//          division-free compare, focal cls loss, smooth-L1 reg loss.
//          32-lane reductions via V_WMMA_F32_16X16X4_F32 (ones-vector trick).
//          Per-block partials (cls,reg,npos) -> workspace (deterministic).
//   pass2: single block; per-image normalize + mean over B -> d_out[2].
// ---------------------------------------------------------------------------

typedef float v2f __attribute__((ext_vector_type(2)));
typedef float v8f __attribute__((ext_vector_type(8)));

#define F_ALPHA 0.2f
#define F_EPS   1e-8f

// Sum a float across all 32 lanes of the wave using the matrix pipe.
// A (16x4 f32): vgpr0 = p (K0 lanes0-15 / K2 lanes16-31), vgpr1 = 0.
// B = ones(4x16)  =>  D[m][n] = p_m + p_{m+16}  (row sums).
// In-lane sum of the 8 D vgprs gives half-sums; one shfl_xor(16) completes.
// EXEC must be all ones: call with every thread of the block active.
__device__ __forceinline__ float wave_sum32_wmma(float p) {
  v2f a; a[0] = p;    a[1] = 0.0f;
  v2f b; b[0] = 1.0f; b[1] = 1.0f;
  v8f c = {0.f, 0.f, 0.f, 0.f, 0.f, 0.f, 0.f, 0.f};
  v8f d = __builtin_amdgcn_wmma_f32_16x16x4_f32(
      /*neg_a=*/false, a, /*neg_b=*/false, b,
      /*c_mod=*/(short)0, c, /*reuse_a=*/false, /*reuse_b=*/false);
  float s = d[0] + d[1] + d[2] + d[3] + d[4] + d[5] + d[6] + d[7];
  s += __shfl_xor(s, 16, 32);
  return s;
}

// One focal-BCE term. Both target cases collapse to a single formula:
//   t=1: 0.2*(1-p)^1.5 * (-log p)     -> q = 1-p
//   t=0: 0.8*( p )^1.5 * (-log(1-p)) -> q = p
__device__ __forceinline__ float focal_term(float praw, bool t1) {
  float p  = fminf(fmaxf(praw, F_EPS), 1.0f - F_EPS);
  float q  = t1 ? (1.0f - p) : p;
  float af = t1 ? F_ALPHA : (1.0f - F_ALPHA);
  return af * q * __builtin_sqrtf(q) * (-__logf(1.0f - q));
}

__global__ __launch_bounds__(256) void focal_pass1(
    const float* __restrict__ cls,   // [B,A,C]
    const float* __restrict__ reg,   // [B,A,4]
    const float* __restrict__ anc,   // [1,A,4]
    const float* __restrict__ ann,   // [B,M,5]
    float* __restrict__ ws,          // [B*BPI,4] partials: cls,reg,npos,0
    int A, int C, int M, int BPI)
{
  const int tid   = threadIdx.x;
  const int b     = blockIdx.x / BPI;
  const int chunk = blockIdx.x % BPI;

  __shared__ __align__(16) float s_ann[256];   // M*5 <= 256
  __shared__ float s_area[64];
  __shared__ float s_red[3 * 8];

  // ---- broadcast this image's annotations to LDS with async b128 copies ----
  const int   annN = M * 5;
  const int   nvec = annN >> 2;                 // # of 16B chunks
  const float* gann = ann + (size_t)b * annN;
  if (tid < nvec) {
    unsigned lds_addr = (unsigned)(unsigned long long)(&s_ann[tid * 4]);
    unsigned long long gaddr = (unsigned long long)(gann + tid * 4);
    asm volatile("global_load_async_to_lds_b128 %0, %1, off"
                 :: "v"(lds_addr), "v"(gaddr) : "memory");
  }
  {
    int tail = annN & 3;                        // generic tail (0 for M=32)
    if (tid < tail) s_ann[(nvec << 2) + tid] = gann[(nvec << 2) + tid];
  }
  asm volatile("s_wait_asynccnt 0" ::: "memory");
  __syncthreads();
  if (tid < M) {
    const float* g = &s_ann[tid * 5];
    s_area[tid] = (g[2] - g[0]) * (g[3] - g[1]);
  }
  __syncthreads();

  // ---- per-anchor work ----
  float cls_acc = 0.f, reg_acc = 0.f, npos_acc = 0.f;
  const int stride = BPI * 256;
  for (int a = chunk * 256 + tid; a < A; a += stride) {
    float4 an = ((const float4*)anc)[a];
    float aw = an.z - an.x, ah = an.w - an.y;
    float acx = an.x + 0.5f * aw, acy = an.y + 0.5f * ah;
    float area_a = aw * ah;

    // argmax of inter/ua without 32 divisions: compare cross-products.
    float b_int = -2.0f, b_ua = 1.0f;
    int   bm = 0;
    for (int m = 0; m < M; ++m) {
      const float* g = &s_ann[m * 5];
      float iw = fminf(an.z, g[2]) - fmaxf(an.x, g[0]);
      float ih = fminf(an.w, g[3]) - fmaxf(an.y, g[1]);
      iw = fmaxf(iw, 0.f); ih = fmaxf(ih, 0.f);
      float inter = iw * ih;
      float ua    = fmaxf(area_a + s_area[m] - inter, F_EPS);
      if (g[4] == -1.0f) { inter = -1.0f; ua = 1.0f; }     // padded GT -> -1
      if (inter * b_ua > b_int * ua) { b_int = inter; b_ua = ua; bm = m; }
    }
    float iou_max = b_int / b_ua;
    bool pos = iou_max >= 0.5f;
    bool neg = iou_max <  0.4f;
    int  lbl = (int)s_ann[bm * 5 + 4];
    float gate = (pos || neg) ? 1.0f : 0.0f;    // ignore band contributes 0

    // classification focal loss over C classes (b128 loads)
    size_t cbase = ((size_t)b * A + a) * (size_t)C;
    const float4* cp = (const float4*)(cls + cbase);
    float csum = 0.f;
    for (int cc = 0; cc < (C >> 2); ++cc) {
      float4 pv = cp[cc];
      int c0 = cc << 2;
      csum += focal_term(pv.x, pos && (c0 + 0 == lbl));
      csum += focal_term(pv.y, pos && (c0 + 1 == lbl));
      csum += focal_term(pv.z, pos && (c0 + 2 == lbl));
      csum += focal_term(pv.w, pos && (c0 + 3 == lbl));
    }
    for (int c = C & ~3; c < C; ++c)            // generic tail (0 for C=20)
      csum += focal_term(cls[cbase + c], pos && (c == lbl));
    cls_acc += gate * csum;

    // regression smooth-L1 (positives only; skips the reg read otherwise)
    if (pos) {
      const float* g = &s_ann[bm * 5];
      float gw = g[2] - g[0], gh = g[3] - g[1];
      float gcx = g[0] + 0.5f * gw, gcy = g[1] + 0.5f * gh;
      gw = fmaxf(gw, 1.0f); gh = fmaxf(gh, 1.0f);
      float t0 = ((gcx - acx) / aw) / 0.1f;
      float t1 = ((gcy - acy) / ah) / 0.1f;
      float t2 = __logf(gw / aw) / 0.2f;
      float t3 = __logf(gh / ah) / 0.2f;
      float4 rg = ((const float4*)reg)[(size_t)b * A + a];
      float d0 = fabsf(t0 - rg.x), d1 = fabsf(t1 - rg.y);
      float d2 = fabsf(t2 - rg.z), d3 = fabsf(t3 - rg.w);
      const float th = 1.0f / 9.0f, hb = 0.5f / 9.0f;
      reg_acc += (d0 <= th ? 4.5f * d0 * d0 : d0 - hb)
               + (d1 <= th ? 4.5f * d1 * d1 : d1 - hb)
               + (d2 <= th ? 4.5f * d2 * d2 : d2 - hb)
               + (d3 <= th ? 4.5f * d3 * d3 : d3 - hb);
      npos_acc += 1.0f;
    }
  }

  // ---- block reduction: wave-level via WMMA, cross-wave via LDS ----
  float wc = wave_sum32_wmma(cls_acc);
  float wr = wave_sum32_wmma(reg_acc);
  float wn = wave_sum32_wmma(npos_acc);
  int wid = tid >> 5, lane = tid & 31;
  if (lane == 0) { s_red[wid*3+0] = wc; s_red[wid*3+1] = wr; s_red[wid*3+2] = wn; }
  __syncthreads();
  if (tid == 0) {
    float tc = 0.f, tr = 0.f, tn = 0.f;
    int nw = blockDim.x >> 5;
    for (int i = 0; i < nw; ++i) { tc += s_red[i*3]; tr += s_red[i*3+1]; tn += s_red[i*3+2]; }
    float* o = ws + (size_t)blockIdx.x * 4;
    o[0] = tc; o[1] = tr; o[2] = tn; o[3] = 0.f;
  }
}

__global__ __launch_bounds__(256) void focal_finalize(
    const float* __restrict__ ws, const float* __restrict__ ann,
    float* __restrict__ out, int B, int BPI, int M)
{
  const int tid = threadIdx.x;
  const int w   = tid >> 5;      // wave w handles image w
  const int lane = tid & 31;

  float pc = 0.f, pr = 0.f, pn = 0.f, pv = 0.f;
  if (w < B) {
    for (int i = lane; i < BPI; i += 32) {
      const float* p = ws + ((size_t)w * BPI + i) * 4;
      pc += p[0]; pr += p[1]; pn += p[2];
    }
    for (int m = lane; m < M; m += 32)
      pv += (ann[((size_t)w * M + m) * 5 + 4] != -1.0f) ? 1.0f : 0.0f;
  }
  // all 256 threads participate (EXEC all ones for WMMA)
  float sc = wave_sum32_wmma(pc);
  float sr = wave_sum32_wmma(pr);
  float sn = wave_sum32_wmma(pn);
  float sv = wave_sum32_wmma(pv);

  __shared__ float s_cl[8], s_rl[8];
  if (lane == 0 && w < 8) {
    float cls_l = sc / fmaxf(sn, 1.0f);
    float reg_l = (sn > 0.0f) ? sr / fmaxf(4.0f * sn, 1.0f) : 0.0f;
    if (!(sv > 0.0f)) { cls_l = 0.0f; reg_l = 0.0f; }
    s_cl[w] = (w < B) ? cls_l : 0.0f;
    s_rl[w] = (w < B) ? reg_l : 0.0f;
  }
  __syncthreads();
  if (tid == 0) {
    float mc = 0.f, mr = 0.f;
    int n = (B < 8) ? B : 8;
    for (int i = 0; i < n; ++i) { mc += s_cl[i]; mr += s_rl[i]; }
    out[0] = mc / (float)B;
    out[1] = mr / (float)B;
  }
}

extern "C" void kernel_launch(void* const* d_in, const int* in_sizes, int n_in,
                              void* d_out, int out_size, void* d_ws, size_t ws_size,
                              hipStream_t stream) {
  const float* cls = (const float*)d_in[0];   // [B,A,C]
  const float* reg = (const float*)d_in[1];   // [B,A,4]
  const float* anc = (const float*)d_in[2];   // [1,A,4]
  const float* ann = (const float*)d_in[3];   // [B,M,5]

  const int A = in_sizes[2] / 4;
  const long long BA = (long long)in_sizes[1] / 4;      // B*A
  const int B = (int)(BA / A);
  const int M = in_sizes[3] / (B * 5);
  const int C = (int)((long long)in_sizes[0] / BA);

  const int BPI = 128;                                  // blocks per image
  float* ws = (float*)d_ws;                             // B*BPI*4 floats = 16 KB

  focal_pass1<<<dim3(B * BPI), dim3(256), 0, stream>>>(
      cls, reg, anc, ann, ws, A, C, M, BPI);
  focal_finalize<<<dim3(1), dim3(256), 0, stream>>>(
      ws, ann, (float*)d_out, B, BPI, M);
}